// RadixAttention_30674656428065
// MI455X (gfx1250) — compile-verified
//
#include <hip/hip_runtime.h>
#include <math.h>

// RadixAttention decode (GQA), fp32, MI455X/gfx1250.
// B=32, HQ=32, HKV=8 (G=4 query heads per kv head), D=128, T=2048.
// HBM-bound (~2 flop/byte): stream K/V once (non-temporal), fp32 WMMA 16x16x4
// for both QK^T and PV, flash-decode online softmax, 8 waves/block merged in LDS.

#define B_    32
#define HQ_   32
#define HKV_  8
#define G_    4
#define D_    128
#define T_    2048
#define NW    8           // waves per block (wave32)
#define SCALING_  0.08838834764831845f
#define LOGIT_CAP_ 30.0f
#define NEG_  -1e30f

typedef float v2f __attribute__((ext_vector_type(2)));
typedef float v8f __attribute__((ext_vector_type(8)));

// D(16x16,f32) = A(16x4,f32) x B(4x16,f32) + C.
// 8-arg pattern: (neg_a, A, neg_b, B, c_mod, C, reuse_a, reuse_b)
static __device__ __forceinline__ v8f wmma4(v2f a, v2f b, v8f c) {
  return __builtin_amdgcn_wmma_f32_16x16x4_f32(false, a, false, b, (short)0, c,
                                               false, false);
}

__global__ __launch_bounds__(256) void radix_decode_wmma(
    const float* __restrict__ q, const float* __restrict__ knew,
    const float* __restrict__ vnew, const float* __restrict__ kbuf,
    const float* __restrict__ vbuf, const int* __restrict__ req_to_token,
    const int* __restrict__ req_pool_indices, const int* __restrict__ seq_lens,
    const int* __restrict__ out_cache_loc, float* __restrict__ out) {
  __shared__ float lds_o[G_ * D_];   // o^T accumulation: [d][g]
  __shared__ float lds_m[NW * 32];
  __shared__ float lds_l[NW * 32];
  __shared__ float lds_L[G_];

  const int tid  = threadIdx.x;
  const int wid  = tid >> 5;
  const int lane = tid & 31;
  const int lm   = lane & 15;        // M / N index within a 16-wide tile
  const bool hi  = lane >= 16;
  const int koff = hi ? 2 : 0;       // K-dim offset held by this lane half

  const int bh = blockIdx.x;
  const int b  = bh / HKV_;
  const int h  = bh % HKV_;

  const int L   = seq_lens[b];
  const int rp  = req_pool_indices[b];
  const int loc = out_cache_loc[b];

  for (int i = tid; i < G_ * D_; i += blockDim.x) lds_o[i] = 0.0f;

  // ---- Preload Q^T in WMMA B-operand layout: rows = d-chunk of 4, cols = query.
  // B 4x16 layout: VGPR0 holds K-rows {0 (lanes0-15), 2 (lanes16-31)}, VGPR1 {1,3}
  // -> per lane the two rows are contiguous d's (koff, koff+1): one aligned b64 load.
  const int qg = (lm < G_) ? lm : 0;                  // clamp garbage query lanes
  const float* qp = q + ((size_t)b * HQ_ + (size_t)(h * G_ + qg)) * D_;
  v2f qb[32];
#pragma unroll
  for (int c = 0; c < 32; ++c) {
    v2f t = *(const v2f*)(qp + 4 * c + koff);
    if (lm >= G_) { t.x = 0.0f; t.y = 0.0f; }         // zero the 12 unused N cols
    qb[c] = t;
  }

  v8f acc[8];                                         // O^T tiles: M=d(16), N=query
#pragma unroll
  for (int dt = 0; dt < 8; ++dt) acc[dt] = (v8f){0, 0, 0, 0, 0, 0, 0, 0};
  float m_run = -INFINITY;
  float l_run = 0.0f;

  const float* knewrow = knew + ((size_t)b * HKV_ + h) * D_;
  const float* vnewrow = vnew + ((size_t)b * HKV_ + h) * D_;

  const int ntiles = (L + 15) >> 4;
  for (int tile = wid; tile < ntiles; tile += NW) {
    const int tbase = tile << 4;
    const int tok   = tbase + lm;                     // A-operand row token (M=lm)
    const int slot  = req_to_token[rp * T_ + tok];
    // new token's KV lives in k/v inputs, not the pool (avoid mutating inputs)
    const float* krow =
        (slot == loc) ? knewrow : (kbuf + ((size_t)slot * HKV_ + h) * D_);

    // ---- S_tile = K_tile x Q^T : C layout M=token, N=query -----------------
    v8f c = (v8f){0, 0, 0, 0, 0, 0, 0, 0};
#pragma unroll
    for (int cc = 0; cc < 32; ++cc) {
      v2f a = __builtin_nontemporal_load((const v2f*)(krow + 4 * cc + koff));
      c = wmma4(a, qb[cc], c);
    }

    // ---- scale, soft-cap, mask; online softmax -----------------------------
    float p[8];
    float mt = -INFINITY;
#pragma unroll
    for (int r = 0; r < 8; ++r) {
      float s = c[r] * SCALING_;
      s = LOGIT_CAP_ * tanhf(s * (1.0f / LOGIT_CAP_));
      const int tr = tbase + r + (hi ? 8 : 0);        // token for C VGPR r
      s = (tr < L) ? s : NEG_;
      p[r] = s;
      mt = fmaxf(mt, s);
    }
    mt = fmaxf(mt, __shfl_xor(mt, 16, 32));           // share max across halves
    const float m_new = fmaxf(m_run, mt);
    const float alpha = __expf(m_run - m_new);
    float ps = 0.0f;
#pragma unroll
    for (int r = 0; r < 8; ++r) {
      p[r] = __expf(p[r] - m_new);                    // masked -> exactly 0
      ps += p[r];
    }
    l_run = l_run * alpha + ps;
    m_run = m_new;
#pragma unroll
    for (int dt = 0; dt < 8; ++dt)
#pragma unroll
      for (int r = 0; r < 8; ++r) acc[dt][r] *= alpha;

    // ---- relayout P (C layout) -> B-operand chunks via half swaps ----------
    float swp[8];
#pragma unroll
    for (int r = 0; r < 8; ++r) swp[r] = __shfl_xor(p[r], 16, 32);
    float bp[8];
    bp[0] = hi ? swp[2] : p[0];   // tc=0 : rows 0 /2
    bp[1] = hi ? swp[3] : p[1];   //        rows 1 /3
    bp[2] = hi ? swp[6] : p[4];   // tc=4 : rows 4 /6
    bp[3] = hi ? swp[7] : p[5];   //        rows 5 /7
    bp[4] = hi ? p[2] : swp[0];   // tc=8 : rows 8 /10
    bp[5] = hi ? p[3] : swp[1];   //        rows 9 /11
    bp[6] = hi ? p[6] : swp[4];   // tc=12: rows 12/14
    bp[7] = hi ? p[7] : swp[5];   //        rows 13/15

    // ---- O^T += V^T x P ----------------------------------------------------
#pragma unroll
    for (int ci = 0; ci < 4; ++ci) {
      const int j0 = ci * 4 + koff;                   // tokens for A K-rows
      const int s0 = __shfl(slot, j0, 32);
      const int s1 = __shfl(slot, j0 + 1, 32);
      const float* vb0 =
          (s0 == loc) ? vnewrow : (vbuf + ((size_t)s0 * HKV_ + h) * D_);
      const float* vb1 =
          (s1 == loc) ? vnewrow : (vbuf + ((size_t)s1 * HKV_ + h) * D_);
      v2f bq;
      bq.x = bp[2 * ci];
      bq.y = bp[2 * ci + 1];
#pragma unroll
      for (int dt = 0; dt < 8; ++dt) {
        v2f av;
        av.x = __builtin_nontemporal_load(vb0 + dt * 16 + lm);
        av.y = __builtin_nontemporal_load(vb1 + dt * 16 + lm);
        acc[dt] = wmma4(av, bq, acc[dt]);
      }
    }
  }

  // ---- cross-wave merge through LDS ----------------------------------------
  const float l_tot = l_run + __shfl_xor(l_run, 16, 32);  // combine token halves
  lds_m[wid * 32 + lane] = m_run;
  lds_l[wid * 32 + lane] = l_tot;
  __syncthreads();

  float M = -INFINITY;
#pragma unroll
  for (int w = 0; w < NW; ++w) M = fmaxf(M, lds_m[w * 32 + lane]);
  float Lt = 0.0f;
#pragma unroll
  for (int w = 0; w < NW; ++w)
    Lt += lds_l[w * 32 + lane] * __expf(lds_m[w * 32 + lane] - M);
  const float f_me = __expf(m_run - M);

  if (lm < G_) {                                       // only real query columns
#pragma unroll
    for (int dt = 0; dt < 8; ++dt)
#pragma unroll
      for (int r = 0; r < 8; ++r) {
        const int d = dt * 16 + r + (hi ? 8 : 0);      // C layout: M=d
        atomicAdd(&lds_o[d * G_ + lm], acc[dt][r] * f_me);
      }
  }
  if (tid < G_) lds_L[tid] = Lt;                       // wave0 lanes 0..3, q=tid
  __syncthreads();

  for (int e = tid; e < G_ * D_; e += blockDim.x) {
    const int g = e >> 7;
    const int d = e & (D_ - 1);
    out[((size_t)b * HQ_ + (size_t)(h * G_ + g)) * D_ + d] =
        lds_o[d * G_ + g] / lds_L[g];
  }
}

extern "C" void kernel_launch(void* const* d_in, const int* in_sizes, int n_in,
                              void* d_out, int out_size, void* d_ws,
                              size_t ws_size, hipStream_t stream) {
  const float* q    = (const float*)d_in[0];
  const float* k    = (const float*)d_in[1];
  const float* v    = (const float*)d_in[2];
  const float* kbuf = (const float*)d_in[3];
  const float* vbuf = (const float*)d_in[4];
  const int* req_to_token     = (const int*)d_in[5];
  const int* req_pool_indices = (const int*)d_in[6];
  const int* seq_lens         = (const int*)d_in[7];
  const int* out_cache_loc    = (const int*)d_in[8];
  float* out = (float*)d_out;

  dim3 grid(B_ * HKV_);
  dim3 block(32 * NW);
  radix_decode_wmma<<<grid, block, 0, stream>>>(
      q, k, v, kbuf, vbuf, req_to_token, req_pool_indices, seq_lens,
      out_cache_loc, out);
}